// HiPPO_11055245820570
// MI455X (gfx1250) — compile-verified
//
#include <hip/hip_runtime.h>
#include <hip/hip_bf16.h>

// HiPPO-LegS scan on MI455X (gfx1250):
//   c_t = c_{t-1} @ dA^T + dB * u_t,  c in (64, 256), 4096 steps, out (L,B,N) fp32.
//
// Strategy:
//  - fp32 WMMA (v_wmma_f32_16x16x4_f32): exact fp32 math; dA ~ I + O(1/4096),
//    so low-precision A/B inputs would corrupt the 4096-step accumulation.
//  - 4 workgroups, each owns a 16-row batch tile (M=16). 16 waves/WG, each wave
//    owns one 16-column N tile of the state.
//  - dA^T B-fragments are constant over time -> loaded once into 64 v2f
//    registers per wave (128 VGPRs), never re-read.
//  - c is exchanged between waves through double-buffered LDS (32 KB of the
//    320 KB WGP LDS); one barrier per step.

typedef __attribute__((ext_vector_type(2))) float v2f;
typedef __attribute__((ext_vector_type(8))) float v8f;

#define L_STEPS 4096
#define BATCH   64
#define NST     256
#define BTILE   16          // batch rows per workgroup (= WMMA M)
#define NWAVES  16          // one wave per 16-column N tile
#define KBLOCKS (NST / 4)   // 64 WMMA K-steps of 4

__global__ __launch_bounds__(NWAVES * 32, 1)
void hippo_scan_kernel(const float* __restrict__ u,    // (L, B)
                       const float* __restrict__ dA,   // (N, N) row-major
                       const float* __restrict__ dB,   // (N,)
                       float* __restrict__ out)        // (L, B, N)
{
    __shared__ float cbuf[2][BTILE][NST];   // 2 * 16 * 256 * 4B = 32 KB
    __shared__ float ubuf[2][BTILE];

    const int wave = threadIdx.x >> 5;      // 0..15  (wave32)
    const int lane = threadIdx.x & 31;
    const int half = lane >> 4;             // 0: lanes 0-15, 1: lanes 16-31
    const int l16  = lane & 15;
    const int b0   = blockIdx.x * BTILE;    // batch tile base
    const int n0   = wave * 16;             // this wave's state columns

    // ---- Load time-invariant B fragments: B[k][n] = dA^T[k][n] = dA[n][k].
    // B 4x16 layout (mirror of A): lane l holds (K = k0 + 2*half + {0,1}, N = n0 + l16).
    v2f Bfrag[KBLOCKS];
    {
        const float* dA_row = dA + (size_t)(n0 + l16) * NST + half * 2;
        #pragma unroll
        for (int kb = 0; kb < KBLOCKS; ++kb) {
            Bfrag[kb] = *(const v2f*)(dA_row + kb * 4);   // global_load_b64
        }
    }
    const float dbv = dB[n0 + l16];

    // ---- Init c_{-1} = 0 and stage u_0.
    for (int i = threadIdx.x; i < BTILE * NST; i += blockDim.x)
        ((float*)cbuf[0])[i] = 0.0f;
    if (threadIdx.x < BTILE)
        ubuf[0][threadIdx.x] = u[b0 + threadIdx.x];
    __syncthreads();

    int cur = 0;
    for (int t = 0; t < L_STEPS; ++t) {
        const int nxt = cur ^ 1;

        // A fragment (16x4 fp32 layout): lane l reads row M=l16,
        // K = kb*4 + 2*half + {0,1}  -> one ds_load_b64 per K-block.
        const float* cA = &cbuf[cur][l16][half * 2];

        v8f acc = {0.f, 0.f, 0.f, 0.f, 0.f, 0.f, 0.f, 0.f};
        #pragma unroll
        for (int kb = 0; kb < KBLOCKS; ++kb) {
            v2f Afrag = *(const v2f*)(cA + kb * 4);       // ds_load_b64
            acc = __builtin_amdgcn_wmma_f32_16x16x4_f32(
                /*neg_a=*/false, Afrag,
                /*neg_b=*/false, Bfrag[kb],
                /*c_mod=*/(short)0, acc,
                /*reuse_a=*/false, /*reuse_b=*/false);
        }

        // C/D layout: VGPR i -> (M = i + 8*half, N = n0 + l16).
        // Input term: acc[i] += dB[n] * u[t, b0 + i + 8*half].
        #pragma unroll
        for (int i = 0; i < 8; ++i)
            acc[i] += dbv * ubuf[cur][half * 8 + i];

        // Coalesced global store of this 16x16 tile of c_t.
        float* orow = out + (((size_t)t * BATCH) + b0 + half * 8) * NST + n0 + l16;
        #pragma unroll
        for (int i = 0; i < 8; ++i)
            orow[(size_t)i * NST] = acc[i];

        // Publish c_t into the other LDS buffer; stage u_{t+1}.
        #pragma unroll
        for (int i = 0; i < 8; ++i)
            cbuf[nxt][half * 8 + i][n0 + l16] = acc[i];
        if (t + 1 < L_STEPS && threadIdx.x < BTILE)
            ubuf[nxt][threadIdx.x] = u[(size_t)(t + 1) * BATCH + b0 + threadIdx.x];

        __syncthreads();   // double buffering -> one barrier per step suffices
        cur = nxt;
    }
}

extern "C" void kernel_launch(void* const* d_in, const int* in_sizes, int n_in,
                              void* d_out, int out_size, void* d_ws, size_t ws_size,
                              hipStream_t stream) {
    const float* u  = (const float*)d_in[0];   // (4096, 64)
    const float* dA = (const float*)d_in[1];   // (256, 256)
    const float* dB = (const float*)d_in[2];   // (256,)
    float* out      = (float*)d_out;           // (4096, 64, 256)

    dim3 grid(BATCH / BTILE);      // 4 workgroups, one per 16-row batch tile
    dim3 block(NWAVES * 32);       // 16 wave32 waves
    hippo_scan_kernel<<<grid, block, 0, stream>>>(u, dA, dB, out);
}